// VectorQuantize_2826088481328
// MI455X (gfx1250) — compile-verified
//
#include <hip/hip_runtime.h>
#include <hip/hip_bf16.h>
#include <math.h>

typedef __attribute__((ext_vector_type(2))) float v2f;
typedef __attribute__((ext_vector_type(8))) float v8f;

#define DIM 64
#define NEMB 1024

// ws layout (floats):
//   [0 .. 1023]      per-block loss partial sums
//   [1024 .. 2047]   counts histogram
//   [2048 .. 3071]   e2[j] = ||embed[:,j]||^2
#define WS_LOSS 0
#define WS_CNT  1024
#define WS_E2   2048

// ---------------- init: e2 + zero counts/loss partials ----------------
__global__ __launch_bounds__(256) void vq_init(const float* __restrict__ embed,
                                               float* __restrict__ ws) {
    int j = blockIdx.x * blockDim.x + threadIdx.x;   // 0..1023
    if (j < NEMB) {
        float s = 0.f;
        #pragma unroll 8
        for (int k = 0; k < DIM; ++k) {
            float e = embed[k * NEMB + j];
            s += e * e;
        }
        ws[WS_E2 + j]  = s;
        ws[WS_CNT + j] = 0.f;
        ws[WS_LOSS + j] = 0.f;
    }
}

// ---------------- main: fused GEMM (WMMA f32) + argmin + gather ----------------
// grid: nrows/128 blocks, 256 threads (8 waves of 32; each wave owns 16 rows)
// Each loop iteration handles TWO 16-code blocks with independent accumulator
// chains to double WMMA ILP (A fragments are register-resident and shared).
__global__ __launch_bounds__(256) void vq_main(const float* __restrict__ x,
                                               const float* __restrict__ embed,
                                               const float* __restrict__ ws,
                                               float* __restrict__ counts,
                                               float* __restrict__ loss_part,
                                               float* __restrict__ out) {
    const float* e2g = ws + WS_E2;
    const int lane   = threadIdx.x & 31;
    const int wave   = threadIdx.x >> 5;
    const int rowBase = blockIdx.x * 128 + wave * 16;
    const int lrow   = lane & 15;       // row (for A) / code column (for B,C)
    const int hi16   = lane >> 4;       // which half of the wave
    const int khalf  = hi16 * 2;        // K sub-pair selector {0,1} vs {2,3}

    // A fragments for all 16 K-steps (K=64), pre-scaled by -2 so acc = -2*f.E
    v2f a[16];
    {
        const float* xr = x + (size_t)(rowBase + lrow) * DIM;
        #pragma unroll
        for (int t = 0; t < 16; ++t) {
            a[t].x = -2.0f * xr[4 * t + khalf];
            a[t].y = -2.0f * xr[4 * t + khalf + 1];
        }
    }

    float minv[8];
    int   mini[8];
    #pragma unroll
    for (int i = 0; i < 8; ++i) { minv[i] = 3.4e38f; mini[i] = 0; }

    // sweep the 1024 codes: 32 iterations x 2 independent 16-code blocks
    for (int b = 0; b < 32; ++b) {
        const int n0 = b * 32 + lrow;                // first code column
        const int n1 = n0 + 16;                      // second code column
        v8f acc0 = {};
        v8f acc1 = {};
        const float* bp0 = embed + (size_t)khalf * NEMB + n0;
        const float* bp1 = bp0 + 16;
        #pragma unroll
        for (int t = 0; t < 16; ++t) {
            v2f b0, b1;
            b0.x = bp0[(size_t)(4 * t) * NEMB];
            b0.y = bp0[(size_t)(4 * t + 1) * NEMB];
            b1.x = bp1[(size_t)(4 * t) * NEMB];
            b1.y = bp1[(size_t)(4 * t + 1) * NEMB];
            acc0 = __builtin_amdgcn_wmma_f32_16x16x4_f32(
                false, a[t], false, b0, (short)0, acc0, false, false);
            acc1 = __builtin_amdgcn_wmma_f32_16x16x4_f32(
                false, a[t], false, b1, (short)0, acc1, false, false);
        }
        const float e0 = e2g[n0];
        const float e1 = e2g[n1];
        #pragma unroll
        for (int i = 0; i < 8; ++i) {
            float d0 = e0 + acc0[i];                 // ||e||^2 - 2 f.e (+const)
            if (d0 < minv[i]) { minv[i] = d0; mini[i] = n0; }
            float d1 = e1 + acc1[i];
            if (d1 < minv[i]) { minv[i] = d1; mini[i] = n1; }
        }
    }

    // cross-lane argmin within each 16-lane half (C/D cols live in lane%16)
    #pragma unroll
    for (int mask = 1; mask <= 8; mask <<= 1) {
        #pragma unroll
        for (int i = 0; i < 8; ++i) {
            float ov = __shfl_xor(minv[i], mask, 16);
            int   oi = __shfl_xor(mini[i], mask, 16);
            if (ov < minv[i] || (ov == minv[i] && oi < mini[i])) {
                minv[i] = ov; mini[i] = oi;
            }
        }
    }

    // gather quantize, straight-through output, loss partials, histogram
    float lsum = 0.f;
    #pragma unroll
    for (int i = 0; i < 8; ++i) {
        const int row = rowBase + i + hi16 * 8;      // C/D: lanes>=16 hold M=i+8
        const int idx = mini[i];                     // uniform within 16-lane half
        #pragma unroll
        for (int kk = 0; kk < 4; ++kk) {
            const int k = lrow + kk * 16;
            float q  = embed[(size_t)k * NEMB + idx];
            float xv = x[(size_t)row * DIM + k];
            out[(size_t)row * DIM + k] = xv + (q - xv);  // quantize_st
            float df = q - xv;
            lsum += df * df;
        }
        if (lrow == 0) atomicAdd(&counts[idx], 1.0f);    // exact: integer-valued
    }

    // deterministic loss reduction: wave -> block -> ws partial
    #pragma unroll
    for (int mask = 1; mask < 32; mask <<= 1)
        lsum += __shfl_xor(lsum, mask, 32);
    __shared__ float sred[8];
    if (lane == 0) sred[wave] = lsum;
    __syncthreads();
    if (threadIdx.x == 0) {
        float s = 0.f;
        #pragma unroll
        for (int w = 0; w < 8; ++w) s += sred[w];
        loss_part[blockIdx.x] = s;
    }
}

// ---------------- finalize: loss mean + perplexity ----------------
__global__ __launch_bounds__(256) void vq_finalize(const float* __restrict__ ws,
                                                   float* __restrict__ scal,
                                                   int nrows) {
    __shared__ float redL[256];
    __shared__ float redH[256];
    const int t = threadIdx.x;
    float sl = 0.f, sh = 0.f;
    for (int j = t; j < 1024; j += 256) {
        sl += ws[WS_LOSS + j];
        float p = ws[WS_CNT + j] / (float)nrows;
        sh += p * logf(p + 1e-10f);
    }
    redL[t] = sl; redH[t] = sh;
    __syncthreads();
    for (int o = 128; o > 0; o >>= 1) {
        if (t < o) { redL[t] += redL[t + o]; redH[t] += redH[t + o]; }
        __syncthreads();
    }
    if (t == 0) {
        scal[0] = redL[0] / ((float)nrows * (float)DIM);  // commitment loss (beta=1)
        scal[1] = expf(-redH[0]);                          // perplexity
    }
}

extern "C" void kernel_launch(void* const* d_in, const int* in_sizes, int n_in,
                              void* d_out, int out_size, void* d_ws, size_t ws_size,
                              hipStream_t stream) {
    const float* x     = (const float*)d_in[0];   // [8,16384,64]
    const float* embed = (const float*)d_in[1];   // [64,1024]
    float* out = (float*)d_out;
    float* ws  = (float*)d_ws;

    const int nrows = in_sizes[0] / DIM;          // 131072

    vq_init<<<(NEMB + 255) / 256, 256, 0, stream>>>(embed, ws);
    vq_main<<<nrows / 128, 256, 0, stream>>>(x, embed, ws,
                                             ws + WS_CNT, ws + WS_LOSS, out);
    vq_finalize<<<1, 256, 0, stream>>>(ws, out + (size_t)nrows * DIM, nrows);
}